// RCNN_37383395344749
// MI455X (gfx1250) — compile-verified
//
#include <hip/hip_runtime.h>
#include <hip/hip_bf16.h>

typedef __attribute__((ext_vector_type(2))) float v2f;
typedef __attribute__((ext_vector_type(8))) float v8f;

#define PP 7
#define BC 512
#define FH 40
#define FW 40
#define NROI 96
#define DIN 25088   /* 512*7*7 */
#define DH 4096
#define KCH 4       /* K-split chunks per GEMM */

// ---------------------------------------------------------------------------
// Kernel 1: ROI crop + AdaptiveMaxPool2d((7,7)) -> X[96, 25088]
// X row layout matches pooled.reshape(N,-1): idx = c*49 + ph*7 + pw
// ---------------------------------------------------------------------------
__global__ void roi_pool_kernel(const float* __restrict__ feat,
                                const float* __restrict__ rois,
                                const int* __restrict__ ridx,
                                float* __restrict__ X) {
  int idx = blockIdx.x * blockDim.x + threadIdx.x;
  const int total = NROI * BC * PP * PP;
  if (idx >= total) return;
  int pw = idx % PP;
  int ph = (idx / PP) % PP;
  int c  = (idx / (PP * PP)) % BC;
  int r  = idx / (PP * PP * BC);

  int x1 = (int)floorf(rois[r * 4 + 0] * (float)FW);
  int y1 = (int)floorf(rois[r * 4 + 1] * (float)FH);
  int x2 = (int)ceilf (rois[r * 4 + 2] * (float)FW);
  int y2 = (int)ceilf (rois[r * 4 + 3] * (float)FH);
  int ww = x2 - x1;
  int hh = y2 - y1;

  int xs = x1 + (pw * ww) / PP;
  int xe = x1 + ((pw + 1) * ww + (PP - 1)) / PP;
  int ys = y1 + (ph * hh) / PP;
  int ye = y1 + ((ph + 1) * hh + (PP - 1)) / PP;
  if (xs < 0) xs = 0; if (ys < 0) ys = 0;
  if (xe > FW) xe = FW; if (ye > FH) ye = FH;

  const int b = ridx[r];
  const float* f = feat + ((size_t)(b * BC + c)) * (FH * FW);
  float m = -3.402823466e+38f;
  for (int y = ys; y < ye; ++y) {
    const float* fr = f + y * FW;
    for (int x = xs; x < xe; ++x) m = fmaxf(m, fr[x]);
  }
  X[(size_t)r * DIN + c * (PP * PP) + ph * PP + pw] = m;
}

// ---------------------------------------------------------------------------
// Kernels 2/4: WMMA f32 GEMM partial.  One wave per (16-wide N stripe,
// K-chunk).  Computes all 6 M-tiles (96 rows) for its stripe using
// v_wmma_f32_16x16x4_f32, writes partial sums to part[chunk][96][4096].
// A 16x4 layout: lanes 0-15 -> K=k,k+1 (VGPR0,1); lanes 16-31 -> K=k+2,k+3.
// B 4x16 layout mirrors A.  C/D: VGPR r -> rows r (lanes 0-15) / r+8 (16-31).
// Inner loop marches two pointers; all row offsets are instruction-immediate
// (max t*16*K*4 = 8,028,160 B < 24-bit signed IOFFSET limit).
// ---------------------------------------------------------------------------
template <int K>
__global__ void __launch_bounds__(32)
gemm_wmma_partial(const float* __restrict__ Xm, const float* __restrict__ Wm,
                  float* __restrict__ part) {
  const int lane = threadIdx.x;         // 0..31
  const int half = lane >> 4;           // 0: K lo pair, 1: K hi pair
  const int l16  = lane & 15;
  const int n    = blockIdx.x * 16 + l16;   // output column
  const int kPerChunk = K / KCH;
  const int k0   = blockIdx.y * kPerChunk;

  v8f acc[6];
#pragma unroll
  for (int t = 0; t < 6; ++t) acc[t] = (v8f){0.f,0.f,0.f,0.f,0.f,0.f,0.f,0.f};

  // Marching pointers: A row base for this lane's M-row 0, B stream base.
  const float* aptr = Xm + (size_t)l16 * K + (k0 + 2 * half);
  const float* bptr = Wm + (size_t)(k0 + 2 * half) * DH + n;

  for (int k = k0; k < k0 + kPerChunk; k += 4) {
    v2f bf;
    bf.x = bptr[0];
    bf.y = bptr[DH];
    if (k + 66 < K)   // uniform condition: prefetch weight stream 64 rows ahead
      __builtin_prefetch(bptr + (size_t)64 * DH, 0, 1);

#pragma unroll
    for (int t = 0; t < 6; ++t) {
      // constant byte offset t*16*K*4 folds into the load's IOFFSET
      v2f af = *(const v2f*)(aptr + (size_t)t * 16 * K);
      acc[t] = __builtin_amdgcn_wmma_f32_16x16x4_f32(
          /*neg_a=*/false, af, /*neg_b=*/false, bf,
          /*c_mod=*/(short)0, acc[t], /*reuse_a=*/false, /*reuse_b=*/false);
    }
    aptr += 4;
    bptr += (size_t)4 * DH;
  }

  float* o = part + (size_t)blockIdx.y * (NROI * DH) + n;
#pragma unroll
  for (int t = 0; t < 6; ++t) {
#pragma unroll
    for (int rrow = 0; rrow < 8; ++rrow) {
      const int m = t * 16 + half * 8 + rrow;
      o[(size_t)m * DH] = acc[t][rrow];   // constant offsets m*DH*4 (<= 1.5MB)
    }
  }
}

// ---------------------------------------------------------------------------
// Kernels 3/5: sum K-chunk partials + bias + ReLU
// ---------------------------------------------------------------------------
__global__ void reduce_bias_relu(const float* __restrict__ part,
                                 const float* __restrict__ bias,
                                 float* __restrict__ out) {
  int idx = blockIdx.x * blockDim.x + threadIdx.x;
  if (idx >= NROI * DH) return;
  int n = idx % DH;
  float s = bias[n];
#pragma unroll
  for (int c = 0; c < KCH; ++c) s += part[(size_t)c * (NROI * DH) + idx];
  out[idx] = fmaxf(s, 0.0f);
}

// ---------------------------------------------------------------------------
// Kernel 6: cls = H2@Wc + bc ; bbox = tile(H2@Wb + bb, 2 copies)
// out layout: cls[96*2] @0, bbox[96*2*4] @192
// ---------------------------------------------------------------------------
__global__ void head_kernel(const float* __restrict__ H2,
                            const float* __restrict__ Wc, const float* __restrict__ bc,
                            const float* __restrict__ Wb, const float* __restrict__ bb,
                            float* __restrict__ out) {
  const int m = blockIdx.x;       // ROI row
  const int tid = threadIdx.x;    // 256 threads
  float s[6] = {0.f, 0.f, 0.f, 0.f, 0.f, 0.f};
  const float* hrow = H2 + (size_t)m * DH;
  for (int k = tid; k < DH; k += 256) {
    float h = hrow[k];
    s[0] += h * Wc[k * 2 + 0];
    s[1] += h * Wc[k * 2 + 1];
    s[2] += h * Wb[k * 4 + 0];
    s[3] += h * Wb[k * 4 + 1];
    s[4] += h * Wb[k * 4 + 2];
    s[5] += h * Wb[k * 4 + 3];
  }
  __shared__ float red[6][256];
#pragma unroll
  for (int j = 0; j < 6; ++j) red[j][tid] = s[j];
  __syncthreads();
  for (int st = 128; st > 0; st >>= 1) {
    if (tid < st) {
#pragma unroll
      for (int j = 0; j < 6; ++j) red[j][tid] += red[j][tid + st];
    }
    __syncthreads();
  }
  if (tid == 0) {
    out[m * 2 + 0] = red[0][0] + bc[0];
    out[m * 2 + 1] = red[1][0] + bc[1];
#pragma unroll
    for (int j = 0; j < 4; ++j) {
      float bv = red[2 + j][0] + bb[j];
      out[192 + m * 8 + j]     = bv;   // bbox copy 0
      out[192 + m * 8 + 4 + j] = bv;   // bbox copy 1
    }
  }
}

// ---------------------------------------------------------------------------
// Kernel 7: attn0 = ones([4,40,40]) at out offset 960
// ---------------------------------------------------------------------------
__global__ void ones_kernel(float* __restrict__ out, int n) {
  int idx = blockIdx.x * blockDim.x + threadIdx.x;
  if (idx < n) out[idx] = 1.0f;
}

// ---------------------------------------------------------------------------
extern "C" void kernel_launch(void* const* d_in, const int* in_sizes, int n_in,
                              void* d_out, int out_size, void* d_ws, size_t ws_size,
                              hipStream_t stream) {
  (void)in_sizes; (void)n_in; (void)out_size; (void)ws_size;
  const float* feat = (const float*)d_in[0];
  const float* rois = (const float*)d_in[1];
  const int*   ridx = (const int*)  d_in[2];
  const float* W1   = (const float*)d_in[3];
  const float* b1   = (const float*)d_in[4];
  const float* W2   = (const float*)d_in[5];
  const float* b2   = (const float*)d_in[6];
  const float* Wc   = (const float*)d_in[7];
  const float* bc   = (const float*)d_in[8];
  const float* Wb   = (const float*)d_in[9];
  const float* bb   = (const float*)d_in[10];
  float* out = (float*)d_out;

  // Workspace layout (floats): X | partials (shared by both GEMMs) | H1 | H2
  float* X  = (float*)d_ws;                 // 96*25088   = 2,408,448
  float* Pt = X  + (size_t)NROI * DIN;      // KCH*96*4096 = 1,572,864
  float* H1 = Pt + (size_t)KCH * NROI * DH; // 96*4096     =   393,216
  float* H2 = H1 + (size_t)NROI * DH;       // 96*4096     =   393,216

  // 1) ROI max pool -> X
  {
    int total = NROI * BC * PP * PP;
    roi_pool_kernel<<<(total + 255) / 256, 256, 0, stream>>>(feat, rois, ridx, X);
  }
  // 2) GEMM1 partials: X[96,25088] @ W1[25088,4096]
  gemm_wmma_partial<DIN><<<dim3(DH / 16, KCH), 32, 0, stream>>>(X, W1, Pt);
  // 3) reduce + b1 + ReLU -> H1
  reduce_bias_relu<<<(NROI * DH) / 256, 256, 0, stream>>>(Pt, b1, H1);
  // 4) GEMM2 partials: H1[96,4096] @ W2[4096,4096]
  gemm_wmma_partial<DH><<<dim3(DH / 16, KCH), 32, 0, stream>>>(H1, W2, Pt);
  // 5) reduce + b2 + ReLU -> H2
  reduce_bias_relu<<<(NROI * DH) / 256, 256, 0, stream>>>(Pt, b2, H2);
  // 6) heads -> cls (offset 0), bbox (offset 192)
  head_kernel<<<NROI, 256, 0, stream>>>(H2, Wc, bc, Wb, bb, out);
  // 7) attn0 ones -> offset 960, 4*40*40 = 6400 elements
  ones_kernel<<<(6400 + 255) / 256, 256, 0, stream>>>(out + 960, 6400);
}